// AttController_Vectorized_19146964205852
// MI455X (gfx1250) — compile-verified
//
#include <hip/hip_runtime.h>
#include <math.h>

// ---------------------------------------------------------------------------
// Attitude PID controller, N independent samples. Pure streaming kernel:
// memory-bound (124 B/sample vs ~90 FLOP/sample -> HBM roofline ~5.3us on
// MI455X at 23.3 TB/s). All global traffic is aligned b128; 4 samples per
// thread; wave32. prev_err / d_filt rows 0,1,2,5 are multiplied by kd==0 in
// the reference and are provably dead -> not loaded (saves 32MB of 156MB).
// Default RT temporal hints so the ~124MB working set stays resident in the
// 192MB L2 across timed graph replays. global_prefetch_b8 issued ahead of
// the streams (gfx1250-specific path); one distinct cacheline per lane so a
// wave issues no duplicate prefetch requests.
// WMMA is deliberately NOT used: per-sample 3x3 matvecs with distinct J per
// sample cannot share a WMMA A/B operand; the kernel is ~1000x below the
// compute roofline, so matrix pipes cannot help a bandwidth-bound stream.
// ---------------------------------------------------------------------------

#define PI_F     3.14159265358979323846f
#define TWO_PI_F (2.0f * PI_F)

__device__ __forceinline__ float clamp_pm(float x, float lim) {
  return fminf(lim, fmaxf(-lim, x));
}

__device__ __forceinline__ void unpack4(const float4 q, float* d) {
  d[0] = q.x; d[1] = q.y; d[2] = q.z; d[3] = q.w;
}

// Shared per-sample math (exact reference semantics).
__device__ __forceinline__ void pid_sample(
    float r0, float r1, float r2,      // ref_rpy
    float m0, float m1, float m2,      // meas_rpy
    float w0, float w1, float w2,      // meas_omegab
    const float* Jr,                   // 9 floats, row-major 3x3
    float gi0, float gi1, float gi2,   // integ rows 0..2 (angle)
    float gi3, float gi4, float gi5,   // integ rows 3..5 (rate)
    float pe3, float pe4,              // prev_err rows 3,4 (only nonzero-kd rows)
    float df3, float df4,              // d_filt rows 3,4
    float& t0, float& t1, float& t2)   // tau out
{
  const float dtA    = 1.0f / 100.0f;
  const float dtR    = 1.0f / 500.0f;
  const float alphaR = (1.0f / 500.0f) / (0.005f + 1.0f / 500.0f);

  float e0 = r0 - m0;
  float e1 = r1 - m1;
  float e2 = r2 - m2;
  // yaw wrap: identical to the reference's two jnp.where selects
  e2 = (e2 >  PI_F) ? e2 - TWO_PI_F : e2;
  e2 = (e2 < -PI_F) ? e2 + TWO_PI_F : e2;

  // angle-loop PID: kd == 0, tau == 0  ->  P + I only
  float wr0 = clamp_pm(6.0f * e0 + 1.0f * (gi0 + e0 * dtA), 10.0f);
  float wr1 = clamp_pm(6.0f * e1 + 1.0f * (gi1 + e1 * dtA), 10.0f);
  float wr2 = clamp_pm(3.0f * e2 + 0.5f * (gi2 + e2 * dtA),  5.0f);

  // rate-loop PID
  float er0 = wr0 - w0, er1 = wr1 - w1, er2 = wr2 - w2;
  float i0 = gi3 + er0 * dtR;
  float i1 = gi4 + er1 * dtR;
  float i2 = gi5 + er2 * dtR;

  float dr0 = (er0 - pe3) / dtR;
  float dr1 = (er1 - pe4) / dtR;
  float dn0 = df3 + alphaR * (dr0 - df3);
  float dn1 = df4 + alphaR * (dr1 - df4);

  float a0 = clamp_pm(0.25f * er0 + 0.5f * i0 + 0.0025f * dn0, 1.0f);
  float a1 = clamp_pm(0.25f * er1 + 0.5f * i1 + 0.0025f * dn1, 1.0f);
  float a2 = clamp_pm(0.12f * er2 + 0.1f * i2,                 0.5f); // RATE_KD[2]==0

  // tau = J*alpha_ref + cross(w, J*w)
  float Jw0 = Jr[0] * w0 + Jr[1] * w1 + Jr[2] * w2;
  float Jw1 = Jr[3] * w0 + Jr[4] * w1 + Jr[5] * w2;
  float Jw2 = Jr[6] * w0 + Jr[7] * w1 + Jr[8] * w2;
  float Ja0 = Jr[0] * a0 + Jr[1] * a1 + Jr[2] * a2;
  float Ja1 = Jr[3] * a0 + Jr[4] * a1 + Jr[5] * a2;
  float Ja2 = Jr[6] * a0 + Jr[7] * a1 + Jr[8] * a2;

  t0 = Ja0 + (w1 * Jw2 - w2 * Jw1);
  t1 = Ja1 + (w2 * Jw0 - w0 * Jw2);
  t2 = Ja2 + (w0 * Jw1 - w1 * Jw0);
}

// Fast path: N % 4 == 0. One thread handles 4 consecutive samples; every
// global access is an aligned float4 (b128).
__global__ __launch_bounds__(256) void att_ctrl_vec4(
    const float* __restrict__ ref,  const float* __restrict__ meas,
    const float* __restrict__ omg,  const float* __restrict__ J,
    const float* __restrict__ integ,const float* __restrict__ prev,
    const float* __restrict__ dfil, float* __restrict__ out, int N)
{
  const int nv = N >> 2;
  const int t  = blockIdx.x * blockDim.x + threadIdx.x;
  if (t >= nv) return;

  // ---- gfx1250 streaming prefetch (global_prefetch_b8), ~400KB ahead ----
  // A wave covers 128 samples. For the 12B/sample streams that is 1536B =
  // 12 cachelines per wave window: prefetch one distinct 128B line per lane,
  // only 12 lanes active -> zero duplicate prefetch requests. The J stream
  // strides 144B/thread (>= one line), so per-lane addresses are already
  // ~1 request per line.
  {
    const int PF   = 8192;                       // float4-groups (= threads) ahead
    const int lane = threadIdx.x & 31;
    const int wt   = (t & ~31) + PF;             // wave-aligned future thread idx
    if (wt + 32 <= nv) {
      const size_t smallBase = (size_t)48 * wt;  // byte offset into 12B/sample streams
      if (lane < 12) {
        __builtin_prefetch((const char*)ref  + smallBase + (size_t)lane * 128, 0, 0);
        __builtin_prefetch((const char*)meas + smallBase + (size_t)lane * 128, 0, 0);
        __builtin_prefetch((const char*)omg  + smallBase + (size_t)lane * 128, 0, 0);
      }
      __builtin_prefetch((const char*)J + (size_t)144 * (t + PF), 0, 0);
    }
  }

  const float4* refv  = reinterpret_cast<const float4*>(ref)  + (size_t)3 * t;
  const float4* measv = reinterpret_cast<const float4*>(meas) + (size_t)3 * t;
  const float4* omgv  = reinterpret_cast<const float4*>(omg)  + (size_t)3 * t;
  const float4* Jv    = reinterpret_cast<const float4*>(J)    + (size_t)9 * t;

  float R[12], M[12], W[12], Jm[36];
#pragma unroll
  for (int k = 0; k < 3; ++k) {
    unpack4(refv[k],  R + 4 * k);
    unpack4(measv[k], M + 4 * k);
    unpack4(omgv[k],  W + 4 * k);
  }
#pragma unroll
  for (int k = 0; k < 9; ++k) unpack4(Jv[k], Jm + 4 * k);

  // SoA state rows (each row base r*N floats is 16B aligned since N%4==0)
  float gI[6][4];
#pragma unroll
  for (int r = 0; r < 6; ++r)
    unpack4(reinterpret_cast<const float4*>(integ + (size_t)r * N)[t], gI[r]);
  float pe3[4], pe4[4], df3[4], df4[4];
  unpack4(reinterpret_cast<const float4*>(prev + (size_t)3 * N)[t], pe3);
  unpack4(reinterpret_cast<const float4*>(prev + (size_t)4 * N)[t], pe4);
  unpack4(reinterpret_cast<const float4*>(dfil + (size_t)3 * N)[t], df3);
  unpack4(reinterpret_cast<const float4*>(dfil + (size_t)4 * N)[t], df4);

  float T[12];
#pragma unroll
  for (int u = 0; u < 4; ++u) {
    pid_sample(R[3*u+0], R[3*u+1], R[3*u+2],
               M[3*u+0], M[3*u+1], M[3*u+2],
               W[3*u+0], W[3*u+1], W[3*u+2],
               Jm + 9 * u,
               gI[0][u], gI[1][u], gI[2][u],
               gI[3][u], gI[4][u], gI[5][u],
               pe3[u], pe4[u], df3[u], df4[u],
               T[3*u+0], T[3*u+1], T[3*u+2]);
  }

  float4* outv = reinterpret_cast<float4*>(out) + (size_t)3 * t;
  outv[0] = make_float4(T[0], T[1], T[2],  T[3]);
  outv[1] = make_float4(T[4], T[5], T[6],  T[7]);
  outv[2] = make_float4(T[8], T[9], T[10], T[11]);
}

// Generic fallback: one sample per thread (used only when N % 4 != 0).
__global__ __launch_bounds__(256) void att_ctrl_scalar(
    const float* __restrict__ ref,  const float* __restrict__ meas,
    const float* __restrict__ omg,  const float* __restrict__ J,
    const float* __restrict__ integ,const float* __restrict__ prev,
    const float* __restrict__ dfil, float* __restrict__ out, int N)
{
  const int i = blockIdx.x * blockDim.x + threadIdx.x;
  if (i >= N) return;

  const float* Jr = J + (size_t)9 * i;
  float Jl[9];
#pragma unroll
  for (int k = 0; k < 9; ++k) Jl[k] = Jr[k];

  float t0, t1, t2;
  pid_sample(ref[3*i+0],  ref[3*i+1],  ref[3*i+2],
             meas[3*i+0], meas[3*i+1], meas[3*i+2],
             omg[3*i+0],  omg[3*i+1],  omg[3*i+2],
             Jl,
             integ[(size_t)0*N+i], integ[(size_t)1*N+i], integ[(size_t)2*N+i],
             integ[(size_t)3*N+i], integ[(size_t)4*N+i], integ[(size_t)5*N+i],
             prev[(size_t)3*N+i],  prev[(size_t)4*N+i],
             dfil[(size_t)3*N+i],  dfil[(size_t)4*N+i],
             t0, t1, t2);

  out[3*i+0] = t0;
  out[3*i+1] = t1;
  out[3*i+2] = t2;
}

extern "C" void kernel_launch(void* const* d_in, const int* in_sizes, int n_in,
                              void* d_out, int out_size, void* d_ws, size_t ws_size,
                              hipStream_t stream) {
  const float* ref   = (const float*)d_in[0];
  const float* meas  = (const float*)d_in[1];
  const float* omg   = (const float*)d_in[2];
  const float* J     = (const float*)d_in[3];
  const float* integ = (const float*)d_in[4];
  const float* prev  = (const float*)d_in[5];
  const float* dfil  = (const float*)d_in[6];
  float* out = (float*)d_out;

  const int N = in_sizes[0] / 3;
  if (N <= 0) return;

  if ((N & 3) == 0) {
    const int nv = N >> 2;
    const int blocks = (nv + 255) / 256;
    att_ctrl_vec4<<<blocks, 256, 0, stream>>>(ref, meas, omg, J, integ, prev,
                                              dfil, out, N);
  } else {
    const int blocks = (N + 255) / 256;
    att_ctrl_scalar<<<blocks, 256, 0, stream>>>(ref, meas, omg, J, integ, prev,
                                                dfil, out, N);
  }
}